// QTT3dQuimb_33054068310492
// MI455X (gfx1250) — compile-verified
//
#include <hip/hip_runtime.h>
#include <hip/hip_bf16.h>

typedef __attribute__((ext_vector_type(2))) float v2f;
typedef __attribute__((ext_vector_type(8))) float v8f;
typedef const __attribute__((address_space(1))) float* gfp;   // force global_load path

#define NSAMP   4096
#define TILE_M  16           // samples per wave
#define SPAD    258          // +2 floats: de-conflict 64-bank LDS, keeps 8B align

__device__ __forceinline__ int qdig(int x, int y, int z, int shift) {
    return (((x >> shift) & 1) << 2) | (((y >> shift) & 1) << 1) | ((z >> shift) & 1);
}

// One TT level as a digit-masked GEMM:
//   vout(16 x ROUT) = sum_j [vin * 1(dig==j)](16 x RIN) @ core[:, j, :](RIN x ROUT)
// via V_WMMA_F32_16X16X4_F32. TBLK independent accumulators per k-step keep the
// matrix pipe fed (breaks the D->C RAW chain) and amortize A/LDS + mask work.
template<int RIN, int ROUT>
__device__ __forceinline__ void level_gemm(const float* __restrict__ Bgen,
                                           const float vin[TILE_M][SPAD],
                                           float       vout[TILE_M][SPAD],
                                           int m, int half, int dig)
{
    gfp B = (gfp)Bgen;
    constexpr int  NT   = (ROUT + 15) / 16;     // 16-wide N tiles
    constexpr int  TBLK = (NT >= 4) ? 4 : 1;    // independent WMMA chains
    constexpr int  NBLK = NT / TBLK;
    constexpr bool FULL = (ROUT % 16) == 0;     // folds away tail predicate

    #pragma unroll 1
    for (int tb = 0; tb < NBLK; ++tb) {
        v8f acc[TBLK];
        #pragma unroll
        for (int u = 0; u < TBLK; ++u)
            acc[u] = (v8f){0.f, 0.f, 0.f, 0.f, 0.f, 0.f, 0.f, 0.f};

        const int  s0     = (tb * TBLK) * 16 + m;   // first column this lane serves
        const bool svalid = FULL || (s0 < ROUT);    // only reachable when TBLK==1

        #pragma unroll 1
        for (int j = 0; j < 8; ++j) {
            // wave-uniform skip of digits absent from this 16-sample tile
            if (!__ballot(dig == j)) continue;
            const float mask = (dig == j) ? 1.0f : 0.0f;
            gfp Bj = B + (size_t)j * ROUT + s0;     // column base for this digit

            #pragma unroll 2
            for (int k0 = 0; k0 < RIN; k0 += 4) {
                const int kA = k0 + 2 * half;       // K pair: lanes 0-15 vs 16-31

                // A fragment (16x4 f32): M = m, K = kA,kA+1 -> one ds_load_b64,
                // one pk_mul mask; shared by all TBLK output tiles.
                v2f a;
                a.x = vin[m][kA    ] * mask;
                a.y = vin[m][kA + 1] * mask;

                // TBLK B fragments: constant u*16 folds into load immediates ->
                // one clause of global_load_b32, then TBLK independent WMMAs.
                #pragma unroll
                for (int u = 0; u < TBLK; ++u) {
                    v2f b;
                    if (svalid) {
                        b.x = Bj[(size_t)(kA    ) * (8 * ROUT) + u * 16];
                        b.y = Bj[(size_t)(kA + 1) * (8 * ROUT) + u * 16];
                    } else {
                        b.x = 0.f; b.y = 0.f;
                    }
                    acc[u] = __builtin_amdgcn_wmma_f32_16x16x4_f32(
                                /*neg_a=*/false, a, /*neg_b=*/false, b,
                                /*c_mod=*/(short)0, acc[u],
                                /*reuse_a=*/false, /*reuse_b=*/false);
                }
            }
        }

        // C/D layout: VGPR vv -> M = vv + 8*half, N = m
        #pragma unroll
        for (int u = 0; u < TBLK; ++u) {
            #pragma unroll
            for (int vv = 0; vv < 8; ++vv)
                vout[vv + 8 * half][(tb * TBLK + u) * 16 + m] = acc[u][vv];
        }
    }
}

__global__ __launch_bounds__(32) void qtt3d_wmma_kernel(
    const float* __restrict__ c0, const float* __restrict__ c1,
    const float* __restrict__ c2, const float* __restrict__ c3,
    const float* __restrict__ c4, const float* __restrict__ c5,
    const float* __restrict__ c6, const float* __restrict__ c7,
    const int*   __restrict__ coords,
    float*       __restrict__ out)
{
    __shared__ float vA[TILE_M][SPAD];   // ping
    __shared__ float vB[TILE_M][SPAD];   // pong

    const int lane = threadIdx.x;             // wave32
    const int m    = lane & 15;               // sample-row this lane serves
    const int half = lane >> 4;               // K-pair selector for A/B fragments
    const int sid  = blockIdx.x * TILE_M + m;

    // Every lane holds the coords of its sample row m (lanes 16-31 mirror 0-15).
    const int x = coords[sid * 3 + 0];
    const int y = coords[sid * 3 + 1];
    const int z = coords[sid * 3 + 2];

    // Warm L2 with the big mid-chain cores (global_prefetch_b8).
    __builtin_prefetch(c3, 0, 1);
    __builtin_prefetch(c4, 0, 1);

    // ---- level 0: v[m][s] = core0[0][d0][s], r_out = 8 ----
    {
        gfp C0 = (gfp)c0;
        const int d0 = qdig(x, y, z, 7);
        if (lane < 16) {
            #pragma unroll
            for (int s = 0; s < 8; ++s)
                vA[m][s] = C0[d0 * 8 + s];
        }
        __syncthreads();   // single-wave WG: S_NOP, orders DS
    }

    // ranks: 1,8,64,256,256,256,64,8,1 ; core l: (rin, 8, rout)
    level_gemm<  8,  64>(c1, vA, vB, m, half, qdig(x, y, z, 6)); __syncthreads();
    level_gemm< 64, 256>(c2, vB, vA, m, half, qdig(x, y, z, 5)); __syncthreads();
    level_gemm<256, 256>(c3, vA, vB, m, half, qdig(x, y, z, 4)); __syncthreads();
    level_gemm<256, 256>(c4, vB, vA, m, half, qdig(x, y, z, 3)); __syncthreads();
    level_gemm<256,  64>(c5, vA, vB, m, half, qdig(x, y, z, 2)); __syncthreads();
    level_gemm< 64,   8>(c6, vB, vA, m, half, qdig(x, y, z, 1)); __syncthreads();
    level_gemm<  8,   1>(c7, vA, vB, m, half, qdig(x, y, z, 0)); __syncthreads();

    // final rank is 1: result sits in column 0
    if (lane < 16)
        out[sid] = vB[m][0];
}

extern "C" void kernel_launch(void* const* d_in, const int* in_sizes, int n_in,
                              void* d_out, int out_size, void* d_ws, size_t ws_size,
                              hipStream_t stream) {
    (void)in_sizes; (void)n_in; (void)d_ws; (void)ws_size; (void)out_size;
    const float* c0 = (const float*)d_in[0];
    const float* c1 = (const float*)d_in[1];
    const float* c2 = (const float*)d_in[2];
    const float* c3 = (const float*)d_in[3];
    const float* c4 = (const float*)d_in[4];
    const float* c5 = (const float*)d_in[5];
    const float* c6 = (const float*)d_in[6];
    const float* c7 = (const float*)d_in[7];
    const int*   coords = (const int*)d_in[8];
    float* out = (float*)d_out;

    const int blocks = NSAMP / TILE_M;   // 256 waves, 1 wave per block
    qtt3d_wmma_kernel<<<blocks, 32, 0, stream>>>(c0, c1, c2, c3, c4, c5, c6, c7,
                                                 coords, out);
}